// GraphMaxPooling_42571715837959
// MI455X (gfx1250) — compile-verified
//
#include <hip/hip_runtime.h>
#include <hip/hip_bf16.h>
#include <stdint.h>

// GraphMaxPooling: CSR segment-max, uniform degree 16, D=32 fp32 features.
// out[i, :] = max_{e in [16i, 16i+16)} x[indices[e], :]
//
// Layout: 8 lanes per node (each lane owns a float4 = 16B of the 128B row),
// 4 nodes per wave32, 32 nodes per 256-thread block.
// Indices for the block (512 ints = 2KB) are staged into LDS with the
// gfx1250 async global->LDS copy (ASYNCcnt) and read back via ds_load_b128.
// All gather/store addresses use 32-bit byte offsets so the backend emits
// GVS-mode global_load_b128 (saddr in SGPRs, one v_lshl_or_b32 per gather).

#define THREADS 256
#define NODES_PER_BLOCK 32            // THREADS / 8
#define DEGREE 16
#define DIM 32
#define ROW_BYTES (DIM * 4)           // 128

__global__ __launch_bounds__(THREADS) void GraphMaxPooling_segmax_kernel(
    const float* __restrict__ x,
    const int*  __restrict__ indices,
    float*      __restrict__ out,
    int n_nodes)
{
    __shared__ int s_idx[NODES_PER_BLOCK * DEGREE];   // 2048 bytes

    const int tid        = threadIdx.x;
    const int node0      = blockIdx.x * NODES_PER_BLOCK;
    const int edge_base  = node0 * DEGREE;            // first edge of this block

    if (node0 + NODES_PER_BLOCK <= n_nodes) {
        // ---- Fast path: stage 512 indices via async global->LDS copy ----
        // Each of the 256 threads moves 8 bytes (2 int32).
        // Generic->LDS address: low 32 bits of the flat pointer are the LDS offset.
        unsigned lds_addr = (unsigned)(uintptr_t)(&s_idx[0]) + (unsigned)tid * 8u;
        unsigned voff     = (unsigned)(tid * 2) * 4u;     // byte offset from saddr
        uint64_t saddr    = (uint64_t)(uintptr_t)(indices + edge_base);
        asm volatile("global_load_async_to_lds_b64 %0, %1, %2"
                     :
                     : "v"(lds_addr), "v"(voff), "s"(saddr)
                     : "memory");
        asm volatile("s_wait_asynccnt 0" ::: "memory");
    } else {
        // ---- Tail path: bounds-checked staging (not hit when 32 | n_nodes) ----
        const int total_edges = n_nodes * DEGREE;
        for (int i = tid; i < NODES_PER_BLOCK * DEGREE; i += THREADS) {
            int e = edge_base + i;
            s_idx[i] = (e < total_edges) ? indices[e] : 0;
        }
    }
    __syncthreads();

    const int local_node = tid >> 3;                  // 0..31
    const int node       = node0 + local_node;
    if (node >= n_nodes) return;

    const unsigned col4 = (unsigned)(tid & 7) * 16u;  // byte offset within a row

    // Pull the 16 neighbor ids with 4x ds_load_b128 (broadcast across the
    // 8 lanes of the node group).
    const int4* nidx4 = (const int4*)(&s_idx[local_node * DEGREE]);
    const int4 i0 = nidx4[0];
    const int4 i1 = nidx4[1];
    const int4 i2 = nidx4[2];
    const int4 i3 = nidx4[3];

    // 32-bit byte offsets: idx*128 | col4 (col4 < 128) -> v_lshl_or_b32,
    // then GVS-mode global_load_b128 off the uniform base pointer.
    const char* __restrict__ xb = (const char*)x;
    float4 v[DEGREE];
    {
        const int ids[DEGREE] = { i0.x, i0.y, i0.z, i0.w,
                                  i1.x, i1.y, i1.z, i1.w,
                                  i2.x, i2.y, i2.z, i2.w,
                                  i3.x, i3.y, i3.z, i3.w };
#pragma unroll
        for (int e = 0; e < DEGREE; ++e) {
            unsigned off = ((unsigned)ids[e] << 7) | col4;
            v[e] = *(const float4*)(xb + off);
        }
    }

    // Linear max chain; backend fuses adjacent pairs into v_max3_f32.
    float4 m = v[0];
#pragma unroll
    for (int e = 1; e < DEGREE; ++e) {
        m.x = fmaxf(m.x, v[e].x);
        m.y = fmaxf(m.y, v[e].y);
        m.z = fmaxf(m.z, v[e].z);
        m.w = fmaxf(m.w, v[e].w);
    }

    unsigned out_off = ((unsigned)node << 7) | col4;
    *(float4*)((char*)out + out_off) = m;
}

extern "C" void kernel_launch(void* const* d_in, const int* in_sizes, int n_in,
                              void* d_out, int out_size, void* d_ws, size_t ws_size,
                              hipStream_t stream)
{
    const float* x       = (const float*)d_in[0];
    const int*   indptr  = (const int*)d_in[1];   // uniform degree; unused
    const int*   indices = (const int*)d_in[2];
    float*       out     = (float*)d_out;
    (void)indptr; (void)n_in; (void)out_size; (void)d_ws; (void)ws_size;

    const int n_nodes = in_sizes[1] - 1;          // indptr has N+1 entries
    const int blocks  = (n_nodes + NODES_PER_BLOCK - 1) / NODES_PER_BLOCK;

    GraphMaxPooling_segmax_kernel<<<blocks, THREADS, 0, stream>>>(
        x, indices, out, n_nodes);
}